// TransformerTGNN_31559419691806
// MI455X (gfx1250) — compile-verified
//
#include <hip/hip_runtime.h>

typedef float v2f __attribute__((ext_vector_type(2)));
typedef float v8f __attribute__((ext_vector_type(8)));

#define NT 1024
#define SENT ((int)0x80000000)

// Single-workgroup hierarchical scan for the TGNN timing state machine.
// Thread t owns the contiguous span [t*CH, (t+1)*CH).
// Phases:
//  1) control partials (segmented up-max, last-cross/ustart/labhi/l_c-event monoids)
//  2) Kogge-Stone scan of control monoids in LDS
//  3) replay: up_hat_seq + per-span alpha affine (P,Q)
//  4) Kogge-Stone affine scan -> alpha at span entry
//  5) replay: alpha_seq + per-span y affine (P,Q)  (A=0 at end, A=1 when frozen)
//  6) Kogge-Stone affine scan -> y at span entry
//  7) replay: y_seq (global), then device fence
//  8) replay: per-end-event loss/cnt contributions (reads y_seq at label events)
//  9) WMMA (v_wmma_f32_16x16x4_f32, ones-matrix) reduction of loss & cnt
__global__ __launch_bounds__(NT)
void tgnn_scan_kernel(const float* __restrict__ a,
                      const float* __restrict__ u,
                      const float* __restrict__ up,
                      const float* __restrict__ label,
                      const float* __restrict__ thres2,
                      float* __restrict__ out, int T)
{
    __shared__ int   ls_cr[NT];
    __shared__ float ls_mx[NT];
    __shared__ int   ls_K[NT], ls_TS[NT], ls_L[NT], ls_S[NT];
    __shared__ float ls_P[NT], ls_Q[NT];

    const int t = threadIdx.x;
    const int CH = (T + NT - 1) / NT;
    const int begin = min(t * CH, T);
    const int end   = min(begin + CH, T);

    float* out_y  = out;          // y_seq
    float* out_al = out + T;      // alpha_seq
    float* out_uh = out + 2 * T;  // up_hat_seq

    const bool  hi0 = (begin > 0) ? (up[begin - 1] >= 0.5f) : false;
    const float u0  = (begin > 0) ? u[begin - 1] : 0.0f;

    // ---------------- Phase 1: control partials ----------------
    {
        int cr = 0; float mx = -3.0e38f;
        int k = SENT, ts = SENT, l = SENT, s = SENT;
        bool hp = hi0; float uprev = u0;
        for (int i = begin; i < end; ++i) {
            float uv = u[i], upv = up[i], labv = label[i];
            bool hi   = upv >= 0.5f;
            bool crs  = hi && !hp;
            bool endf = (!hi) && hp;
            if (uprev == 0.0f && uv == 1.0f) ts = i;          // utterance true start
            if (crs) { cr = 1; mx = upv; k = i; }             // run start: reset seg-max
            else     { mx = fmaxf(mx, upv); }
            if (labv >= 0.5f) {                                // l_c events (ref order: lab then end)
                if (hi) { s = (i << 2) | 2; l = i; }          // SQ event + label_frame
                else    { s = (i << 2) | 1; }                 // -1 event
            }
            if (endf) s = (i << 2);                            // l_c reset to 0
            hp = hi; uprev = uv;
        }
        ls_cr[t] = cr; ls_mx[t] = mx;
        ls_K[t] = k; ls_TS[t] = ts; ls_L[t] = l; ls_S[t] = s;
    }
    __syncthreads();

    // ---------------- Phase 2: Kogge-Stone control scan ----------------
    for (int d = 1; d < NT; d <<= 1) {
        int cr = 0, k = SENT, ts = SENT, l = SENT, s = SENT; float mx = -3.0e38f;
        const bool act = (t >= d);
        if (act) { cr = ls_cr[t-d]; mx = ls_mx[t-d]; k = ls_K[t-d];
                   ts = ls_TS[t-d]; l = ls_L[t-d]; s = ls_S[t-d]; }
        __syncthreads();
        if (act) {
            if (!ls_cr[t]) { ls_mx[t] = fmaxf(mx, ls_mx[t]); ls_cr[t] = cr; }
            if (ls_K[t]  == SENT) ls_K[t]  = k;
            if (ls_TS[t] == SENT) ls_TS[t] = ts;
            if (ls_L[t]  == SENT) ls_L[t]  = l;
            if (ls_S[t]  == SENT) ls_S[t]  = s;
        }
        __syncthreads();
    }

    // exclusive control state entering this span (arrays stay resident; P/Q get reused)
    float upmax_in = 0.0f; int K_in = 0, TS_in = 0, L_in = 0, S_in = 0;
    if (t > 0) {
        upmax_in = ls_cr[t-1] ? ls_mx[t-1] : fmaxf(0.0f, ls_mx[t-1]);
        if (ls_K[t-1]  != SENT) K_in  = ls_K[t-1];
        if (ls_TS[t-1] != SENT) TS_in = ls_TS[t-1];
        if (ls_L[t-1]  != SENT) L_in  = ls_L[t-1];
        if (ls_S[t-1]  != SENT) S_in  = ls_S[t-1];
    }
    __syncthreads();

    // ---------------- Phase 3: up_hat + alpha affine partial ----------------
    {
        float um = upmax_in; bool hp = hi0;
        float P = 1.0f, Q = 0.0f;
        for (int i = begin; i < end; ++i) {
            float upv = up[i];
            bool hi  = upv >= 0.5f;
            bool crs = hi && !hp;
            if (crs) um = upv; else um = fmaxf(um, upv);
            float uh = hi ? um : upv;
            out_uh[i] = uh;
            float B = (1.0f - uh) * a[i];
            P = uh * P;
            Q = uh * Q + B;
            hp = hi;
        }
        ls_P[t] = P; ls_Q[t] = Q;
    }
    __syncthreads();

    // ---------------- Phase 4: affine scan (alpha) ----------------
    for (int d = 1; d < NT; d <<= 1) {
        float p = 1.0f, q = 0.0f; const bool act = (t >= d);
        if (act) { p = ls_P[t-d]; q = ls_Q[t-d]; }
        __syncthreads();
        if (act) { float P2 = ls_P[t], Q2 = ls_Q[t]; ls_P[t] = P2 * p; ls_Q[t] = P2 * q + Q2; }
        __syncthreads();
    }
    const float alpha_in = (t > 0) ? ls_Q[t-1] : 0.0f;   // alpha_{-1} = 0
    __syncthreads();

    // ---------------- Phase 5: alpha_seq + y affine partial ----------------
    {
        float um = upmax_in; bool hp = hi0; int K = K_in;
        float alpha = alpha_in;
        float P = 1.0f, Q = 0.0f;
        for (int i = begin; i < end; ++i) {
            float upv = up[i];
            bool hi   = upv >= 0.5f;
            bool crs  = hi && !hp;
            bool endf = (!hi) && hp;
            if (crs) { um = upv; K = i; } else um = fmaxf(um, upv);
            float uh = hi ? um : upv;
            alpha = uh * alpha + (1.0f - uh) * a[i];
            out_al[i] = alpha;
            bool frz = hi && (K > 0) && ((i - K) > 20);
            float Ay, By;
            if (endf)     { Ay = 0.0f;        By = 0.0f; }        // y reset at end
            else if (frz) { Ay = 1.0f;        By = 0.0f; }        // freeze
            else          { Ay = 1.0f - alpha; By = alpha * uh; } // leaky update
            P = Ay * P; Q = Ay * Q + By;
            hp = hi;
        }
        ls_P[t] = P; ls_Q[t] = Q;
    }
    __syncthreads();

    // ---------------- Phase 6: affine scan (y) ----------------
    for (int d = 1; d < NT; d <<= 1) {
        float p = 1.0f, q = 0.0f; const bool act = (t >= d);
        if (act) { p = ls_P[t-d]; q = ls_Q[t-d]; }
        __syncthreads();
        if (act) { float P2 = ls_P[t], Q2 = ls_Q[t]; ls_P[t] = P2 * p; ls_Q[t] = P2 * q + Q2; }
        __syncthreads();
    }
    const float y_in = (t > 0) ? ls_Q[t-1] : 0.0f;       // y_{-1} = 0
    __syncthreads();

    // ---------------- Phase 7: y_seq replay ----------------
    {
        float um = upmax_in; bool hp = hi0; int K = K_in;
        float alpha = alpha_in; float y = y_in;
        for (int i = begin; i < end; ++i) {
            float upv = up[i];
            bool hi   = upv >= 0.5f;
            bool crs  = hi && !hp;
            bool endf = (!hi) && hp;
            if (crs) { um = upv; K = i; } else um = fmaxf(um, upv);
            float uh = hi ? um : upv;
            alpha = uh * alpha + (1.0f - uh) * a[i];
            bool frz = hi && (K > 0) && ((i - K) > 20);
            float yn;
            if (endf)     yn = 0.0f;
            else if (frz) yn = y;
            else          yn = alpha * uh + (1.0f - alpha) * y;
            out_y[i] = yn;
            y = yn; hp = hi;
        }
    }
    __threadfence();     // make y_seq globally visible before cross-span reads
    __syncthreads();

    // ---------------- Phase 8: loss/cnt contributions ----------------
    {
        float um = upmax_in; bool hp = hi0; float uprev = u0;
        int K = K_in, TS = TS_in, L = L_in, S = S_in;
        float alpha = alpha_in, y = y_in;
        float lossP = 0.0f, cntP = 0.0f;
        for (int i = begin; i < end; ++i) {
            float uv = u[i], upv = up[i], labv = label[i], av = a[i];
            bool hi   = upv >= 0.5f;
            bool crs  = hi && !hp;
            bool endf = (!hi) && hp;
            if (uprev == 0.0f && uv == 1.0f) TS = i;
            if (crs) { um = upv; K = i; } else um = fmaxf(um, upv);
            float uh = hi ? um : upv;
            alpha = uh * alpha + (1.0f - uh) * av;
            // label events (ref steps 10-11) precede end handling
            if (labv >= 0.5f) {
                if (hi) { S = (i << 2) | 2; L = i; }
                else    { S = (i << 2) | 1; }
            }
            if (endf) {
                int styp = S & 3, sidx = S >> 2;
                float lc;
                if (styp == 0)      lc = 0.0f;
                else if (styp == 1) lc = -1.0f;
                else { float dy = out_y[sidx] - label[sidx]; lc = dy * dy; }
                int dur = L - TS + 1;
                bool good = (dur >= 3) && (dur <= 40);
                if (lc != -1.0f && lc != 0.0f) {
                    if (good) { lossP += lc; cntP += 1.0f; }          // case2 (RATE=1)
                } else if (lc == 0.0f && y >= 0.5f) {                 // case3 uses y_pre
                    int id = i - K; id = id < 20 ? id : 19; id = id < 0 ? 0 : id;
                    float d2 = y - thres2[id];
                    lossP += d2 * d2; cntP += 1.0f;
                }
                S = (i << 2);                                          // l_c -> 0
            }
            bool frz = hi && (K > 0) && ((i - K) > 20);
            float yn;
            if (endf)     yn = 0.0f;
            else if (frz) yn = y;
            else          yn = alpha * uh + (1.0f - alpha) * y;
            y = yn; hp = hi; uprev = uv;
        }
        ls_P[t] = lossP; ls_Q[t] = cntP;
    }
    __syncthreads();

    // ---------------- Phase 9: WMMA reductions (wave 0, EXEC all ones) ----------------
    // D = ones(16x4) @ B(4x16) + C gives column sums of 64 fp32 values per
    // v_wmma_f32_16x16x4_f32; the total is layout-agnostic: sum(D) = 16 * sum(B).
    if (t < 32) {
        v2f ones = { 1.0f, 1.0f };
        {
            v8f acc = {};
            #pragma unroll
            for (int b = 0; b < NT / 64; ++b) {
                v2f dat = { ls_P[b * 64 + t * 2], ls_P[b * 64 + t * 2 + 1] };
                acc = __builtin_amdgcn_wmma_f32_16x16x4_f32(
                          false, ones, false, dat, (short)0, acc, false, false);
            }
            float s = 0.0f;
            #pragma unroll
            for (int k = 0; k < 8; ++k) s += acc[k];
            for (int off = 16; off > 0; off >>= 1) s += __shfl_xor(s, off, 32);
            if (t == 0) out[3 * T] = s * 0.0625f;        // loss
        }
        {
            v8f acc = {};
            #pragma unroll
            for (int b = 0; b < NT / 64; ++b) {
                v2f dat = { ls_Q[b * 64 + t * 2], ls_Q[b * 64 + t * 2 + 1] };
                acc = __builtin_amdgcn_wmma_f32_16x16x4_f32(
                          false, ones, false, dat, (short)0, acc, false, false);
            }
            float s = 0.0f;
            #pragma unroll
            for (int k = 0; k < 8; ++k) s += acc[k];
            for (int off = 16; off > 0; off >>= 1) s += __shfl_xor(s, off, 32);
            if (t == 0) out[3 * T + 1] = s * 0.0625f;    // cnt
        }
    }
}

extern "C" void kernel_launch(void* const* d_in, const int* in_sizes, int n_in,
                              void* d_out, int out_size, void* d_ws, size_t ws_size,
                              hipStream_t stream) {
    const float* a      = (const float*)d_in[0];
    const float* u      = (const float*)d_in[1];
    const float* up     = (const float*)d_in[2];
    const float* label  = (const float*)d_in[3];
    const float* thres2 = (const float*)d_in[4];
    float* out = (float*)d_out;
    const int T = in_sizes[0];
    tgnn_scan_kernel<<<1, NT, 0, stream>>>(a, u, up, label, thres2, out, T);
}